// CorrespondenceContrastiveLoss_39977555591343
// MI455X (gfx1250) — compile-verified
//
#include <hip/hip_runtime.h>

typedef float v2f __attribute__((ext_vector_type(2)));
typedef float v8f __attribute__((ext_vector_type(8)));

#define CROP_X 100
#define CROP_Y 88
#define CROP_Z 80
#define CH_STRIDE ((size_t)(CROP_X * CROP_Y * CROP_Z)) // 704000 elems between channels
#define NCH 32

// One wave32 handles a tile of 16 points.
// Lane L: half = L>>4, prow = L&15 -> supplies channels {4k+2*half, 4k+2*half+1}
// of point (tile_base + prow) for k = 0..7.
// For G = D * D^T with V_WMMA_F32_16X16X4_F32, the A(16x4) and B(4x16) operand
// register layouts coincide element-for-element, so the same register pair is
// passed as both A and B. diag(G) = per-point squared distance.
__global__ __launch_bounds__(64) void ccl_main(
    const float* __restrict__ fix, const float* __restrict__ mov,
    const int* __restrict__ fpts, const int* __restrict__ ppts,
    const int* __restrict__ npts, float* __restrict__ out, int N)
{
    const int tid  = blockIdx.x * blockDim.x + threadIdx.x;
    const int wave = tid >> 5;
    const int lane = tid & 31;
    const int half = lane >> 4;
    const int prow = lane & 15;
    const int n0   = wave * 16;                 // first point of this tile
    int p = n0 + prow;
    if (p >= N) p = N - 1;                      // clamp loads; EXEC must stay all-1s for WMMA

    // gather coordinates (pts % crop_half_size)
    const int fx = fpts[3 * p + 0] % CROP_X;
    const int fy = fpts[3 * p + 1] % CROP_Y;
    const int fz = fpts[3 * p + 2] % CROP_Z;
    const int px = ppts[3 * p + 0] % CROP_X;
    const int py = ppts[3 * p + 1] % CROP_Y;
    const int pz = ppts[3 * p + 2] % CROP_Z;
    const int nx = npts[3 * p + 0] % CROP_X;
    const int ny = npts[3 * p + 1] % CROP_Y;
    const int nz = npts[3 * p + 2] % CROP_Z;
    const size_t offF = ((size_t)fx * CROP_Y + fy) * CROP_Z + fz;
    const size_t offP = ((size_t)px * CROP_Y + py) * CROP_Z + pz;
    const size_t offN = ((size_t)nx * CROP_Y + ny) * CROP_Z + nz;

    // Issue all 48 gathered loads up front (independent -> deep LOADcnt pipeline),
    // then form diff operands.
    v2f dP[8], dN[8];
#pragma unroll
    for (int k = 0; k < 8; ++k) {
        const size_t c0 = (size_t)(4 * k + 2 * half);
        const float ff0 = fix[c0 * CH_STRIDE + offF];
        const float ff1 = fix[(c0 + 1) * CH_STRIDE + offF];
        const float mp0 = mov[c0 * CH_STRIDE + offP];
        const float mp1 = mov[(c0 + 1) * CH_STRIDE + offP];
        const float mn0 = mov[c0 * CH_STRIDE + offN];
        const float mn1 = mov[(c0 + 1) * CH_STRIDE + offN];
        dP[k] = (v2f){ff0 - mp0, ff1 - mp1};
        dN[k] = (v2f){ff0 - mn0, ff1 - mn1};
    }

    v8f accP = {};
    v8f accN = {};
#pragma unroll
    for (int k = 0; k < 8; ++k) {
        // two independent accumulation chains -> interleave to hide WMMA latency
        accP = __builtin_amdgcn_wmma_f32_16x16x4_f32(
            false, dP[k], false, dP[k], (short)0, accP, false, false);
        accN = __builtin_amdgcn_wmma_f32_16x16x4_f32(
            false, dN[k], false, dN[k], (short)0, accN, false, false);
    }

    // diag(n,n): n<8 -> VGPR n @ lane n ; n>=8 -> VGPR n-8 @ lane n+16
    const int sel = (lane < 8) ? lane : (lane - 24);
    float d2p = accP[0];
    float d2n = accN[0];
#pragma unroll
    for (int i = 1; i < 8; ++i) {
        d2p = (sel == i) ? accP[i] : d2p;
        d2n = (sel == i) ? accN[i] : d2n;
    }

    const bool writer = (lane < 8) || (lane >= 24);
    const int  n      = n0 + ((lane < 8) ? lane : (lane - 16));
    if (writer && n < N) {
        out[1 + n]     = __builtin_sqrtf(d2p); // pos_dis
        out[1 + N + n] = __builtin_sqrtf(d2n); // neg_dis
    }
}

// Deterministic single-block reduction of the scalar loss from the distance
// arrays already stored in d_out (pos_d2 reconstructed as pos_dis^2).
__global__ __launch_bounds__(256) void ccl_reduce(
    float* __restrict__ out, const float* __restrict__ marginp, int N)
{
    __shared__ float sh[256];
    const float m = *marginp;
    float acc = 0.0f;
    for (int i = (int)threadIdx.x; i < N; i += 256) {
        const float pd = out[1 + i];
        const float nd = out[1 + N + i];
        const float t  = fmaxf(0.0f, m - nd);
        acc += pd * pd + t * t;
    }
    sh[threadIdx.x] = acc;
    __syncthreads();
    for (int s = 128; s > 0; s >>= 1) {
        if ((int)threadIdx.x < s) sh[threadIdx.x] += sh[threadIdx.x + s];
        __syncthreads();
    }
    if (threadIdx.x == 0) {
        // loss = (sum_pos_d2 + sum_neg_term) / (2 * 2N) * 100
        out[0] = sh[0] * (100.0f / (4.0f * (float)N));
    }
}

extern "C" void kernel_launch(void* const* d_in, const int* in_sizes, int n_in,
                              void* d_out, int out_size, void* d_ws, size_t ws_size,
                              hipStream_t stream) {
    (void)n_in; (void)d_ws; (void)ws_size; (void)out_size;
    const float* fix    = (const float*)d_in[0];
    const float* mov    = (const float*)d_in[1];
    const int*   fpts   = (const int*)d_in[2];
    const int*   ppts   = (const int*)d_in[3];
    const int*   npts   = (const int*)d_in[4];
    const float* margin = (const float*)d_in[5];
    float*       out    = (float*)d_out;

    const int N       = in_sizes[2] / 3;        // 4096 points
    const int waves   = (N + 15) / 16;          // 16 points per wave32
    const int threads = waves * 32;
    const int block   = 64;                     // 2 waves/block, spread across WGPs
    const int grid    = (threads + block - 1) / block;

    ccl_main<<<grid, block, 0, stream>>>(fix, mov, fpts, ppts, npts, out, N);
    ccl_reduce<<<1, 256, 0, stream>>>(out, margin, N);
}